// PNAAggregator_9036611191395
// MI455X (gfx1250) — compile-verified
//
#include <hip/hip_runtime.h>
#include <math.h>

// Problem shape (fixed by the reference)
#define MDIM 2048   // rows of adj / output
#define NDIM 2048   // reduction dim of aggregation
#define KDIM 64     // feature dim
#define ODIM 128    // output dim
#define JDIM 256    // 4*KDIM = width of "combined"

typedef __attribute__((ext_vector_type(2))) float v2f;
typedef __attribute__((ext_vector_type(4))) float v4f;
typedef __attribute__((ext_vector_type(8))) float v8f;

// ---------------------------------------------------------------------------
// Stage 1: sparse-aware PNA aggregation.
// One wave32 per row m. Stream adj[m,:] as b128 loads, ballot the nonzero
// lanes into a uniform mask, and only touch features rows for set bits
// (~2% density -> ~41 feature rows per adj row). Each lane owns 2 of the
// 64 k-columns. Exactly reproduces mean / max / min / std(ddof=1) over the
// full N axis (zero products included via the nonzero count).
// ---------------------------------------------------------------------------
__global__ __launch_bounds__(256) void pna_aggregate(
    const float* __restrict__ adj, const float* __restrict__ feat,
    float* __restrict__ combined)
{
    const int lane = threadIdx.x & 31;
    const int wave = threadIdx.x >> 5;
    const int m = blockIdx.x * 8 + wave;   // 256 blocks * 8 waves = 2048 rows

    float s1a = 0.f, s1b = 0.f, s2a = 0.f, s2b = 0.f;
    float mxa = -INFINITY, mxb = -INFINITY;
    float mna =  INFINITY, mnb =  INFINITY;
    unsigned cnt = 0;   // wave-uniform count of nonzero adj entries

    const float* arow = adj + (size_t)m * NDIM;
    for (int n0 = 0; n0 < NDIM; n0 += 128) {
        // lane holds adj[m, n0+4*lane .. n0+4*lane+3]
        const v4f av = *reinterpret_cast<const v4f*>(arow + n0 + 4 * lane);
#pragma unroll
        for (int c = 0; c < 4; ++c) {
            unsigned mask = __builtin_amdgcn_ballot_w32(av[c] != 0.0f);
            cnt += (unsigned)__builtin_popcount(mask);
            while (mask) {                       // uniform sparse scan
                const int bit = __builtin_ctz(mask);
                mask &= mask - 1u;
                const int n = n0 + 4 * bit + c;  // uniform nonzero column
                const float f0 = feat[(size_t)n * KDIM + lane];
                const float f1 = feat[(size_t)n * KDIM + 32 + lane];
                s1a += f0;                 s1b += f1;
                s2a = fmaf(f0, f0, s2a);   s2b = fmaf(f1, f1, s2b);
                mxa = fmaxf(mxa, f0);      mxb = fmaxf(mxb, f1);
                mna = fminf(mna, f0);      mnb = fminf(mnb, f1);
            }
        }
    }

    // Zero products participate in max/min whenever any adj[m,n]==0.
    if (cnt < (unsigned)NDIM) {
        mxa = fmaxf(mxa, 0.f); mxb = fmaxf(mxb, 0.f);
        mna = fminf(mna, 0.f); mnb = fminf(mnb, 0.f);
    }

    const float invN   = 1.0f / (float)NDIM;
    const float invNm1 = 1.0f / (float)(NDIM - 1);
    const float mean_a = s1a * invN, mean_b = s1b * invN;
    const float var_a = fmaxf((s2a - s1a * s1a * invN) * invNm1, 0.f);
    const float var_b = fmaxf((s2b - s1b * s1b * invN) * invNm1, 0.f);

    float* crow = combined + (size_t)m * JDIM;   // [mean | max | min | std]
    crow[lane]       = mean_a;         crow[32 + lane]  = mean_b;
    crow[64 + lane]  = mxa;            crow[96 + lane]  = mxb;
    crow[128 + lane] = mna;            crow[160 + lane] = mnb;
    crow[192 + lane] = sqrtf(var_a);   crow[224 + lane] = sqrtf(var_b);
}

// ---------------------------------------------------------------------------
// Stage 2: out = tanh(combined(2048x256) @ W(256x128) + b) via
// V_WMMA_F32_16X16X4_F32. One wave per 16x16 output tile, 64 chained WMMAs
// over K=256. fp32 A/B keeps exact reference numerics.
// A 16x4 layout: VGPR0 = K {0|2} across lane halves, VGPR1 = K {1|3};
// C/D layout: N = lane&15, M = vgpr + 8*(lane>=16).
// ---------------------------------------------------------------------------
__global__ __launch_bounds__(256) void pna_gemm_tanh(
    const float* __restrict__ combined, const float* __restrict__ W,
    const float* __restrict__ bias, float* __restrict__ out)
{
    const int lane = threadIdx.x & 31;
    const int wave = threadIdx.x >> 5;
    const int tile = blockIdx.x * 8 + wave;   // 128 blocks * 8 waves = 1024 tiles
    const int m0 = (tile >> 3) * 16;          // 128 row tiles
    const int o0 = (tile & 7) * 16;           // 8 col tiles

    const int mrow  = m0 + (lane & 15);
    const int ocol  = o0 + (lane & 15);
    const int khalf = (lane >> 4) * 2;        // 0 for lanes 0-15, 2 for 16-31

    v8f c = {0.f, 0.f, 0.f, 0.f, 0.f, 0.f, 0.f, 0.f};
    const float* arow = combined + (size_t)mrow * JDIM + khalf;

#pragma unroll 4
    for (int kk = 0; kk < JDIM; kk += 4) {
        v2f a, b;
        a.x = arow[kk];
        a.y = arow[kk + 1];
        b.x = W[(size_t)(kk + khalf) * ODIM + ocol];
        b.y = W[(size_t)(kk + khalf + 1) * ODIM + ocol];
        // (neg_a, A, neg_b, B, c_mod, C, reuse_a, reuse_b)
        c = __builtin_amdgcn_wmma_f32_16x16x4_f32(false, a, false, b,
                                                  (short)0, c, false, false);
    }

    const float bv = bias[ocol];
#pragma unroll
    for (int v = 0; v < 8; ++v) {
        const int mm = m0 + v + ((lane >> 4) << 3);
        out[(size_t)mm * ODIM + ocol] = tanhf(c[v] + bv);
    }
}

extern "C" void kernel_launch(void* const* d_in, const int* in_sizes, int n_in,
                              void* d_out, int out_size, void* d_ws, size_t ws_size,
                              hipStream_t stream) {
    const float* adj  = (const float*)d_in[0];   // (2048, 2048)
    const float* feat = (const float*)d_in[1];   // (2048, 64)
    const float* W    = (const float*)d_in[2];   // (256, 128)
    const float* bias = (const float*)d_in[3];   // (128,)
    float* out = (float*)d_out;                  // (2048, 128)
    float* combined = (float*)d_ws;              // (2048, 256) scratch, 2 MB

    pna_aggregate<<<MDIM / 8, 256, 0, stream>>>(adj, feat, combined);
    pna_gemm_tanh<<<(MDIM / 16) * (ODIM / 16) / 8, 256, 0, stream>>>(
        combined, W, bias, out);
}